// RenyiMutualInformation_87917980549868
// MI455X (gfx1250) — compile-verified
//
#include <hip/hip_runtime.h>
#include <math.h>

typedef __attribute__((ext_vector_type(2))) float v2f;
typedef __attribute__((ext_vector_type(8))) float v8f;

#define NB 512
#define NT 256
#define WPB (NT / 32)

#define ALPHA_F   2.0f
#define NUM_STEPS 8
#define ADAM_LR   1e-3f
#define ADAM_B1   0.9f
#define ADAM_B2   0.999f
#define ADAM_EPS  1e-8f

// workspace layout (float offsets)
#define WS_P    0      // params: w1[64] b1[4] w2[4] b2[1] (pad to 80)
#define WS_M    80
#define WS_V    160
#define WS_G    240
#define WS_SC   320    // lse1, lse2
#define WS_PART 336    // NB * 4 floats: (max1,sum1,max2,sum2) per block

// ---------------- logsumexp helpers ----------------
__device__ __forceinline__ void lse_combine(float& m, float& s, float om, float os) {
  float nm = fmaxf(m, om);
  if (nm == -INFINITY) { m = nm; s = 0.0f; return; }
  s = s * expf(m - nm) + os * expf(om - nm);
  m = nm;
}

__device__ __forceinline__ void warp_reduce_lse(float& m, float& s) {
#pragma unroll
  for (int o = 16; o > 0; o >>= 1) {
    float om = __shfl_xor(m, o, 32);
    float os = __shfl_xor(s, o, 32);
    lse_combine(m, s, om, os);
  }
}

// ---------------- WMMA layer-1 (16 samples/tile, K=16 via 4x K=4) ----------------
// A = w1 padded 16x16 (rows 0..3 live), B = 16 features x 16 samples.
// D[m][n] = sum_k w1[m][k] * feat[n][k]  -> lane n (0..15) holds z1[n][m] in C[m].
__device__ __forceinline__ v8f mlp_l1_wmma(const v2f a[4], v2f b0, v2f b1, v2f b2, v2f b3) {
  v8f c = {0.f, 0.f, 0.f, 0.f, 0.f, 0.f, 0.f, 0.f};
  c = __builtin_amdgcn_wmma_f32_16x16x4_f32(false, a[0], false, b0, (short)0, c, false, false);
  c = __builtin_amdgcn_wmma_f32_16x16x4_f32(false, a[1], false, b1, (short)0, c, false, false);
  c = __builtin_amdgcn_wmma_f32_16x16x4_f32(false, a[2], false, b2, (short)0, c, false, false);
  c = __builtin_amdgcn_wmma_f32_16x16x4_f32(false, a[3], false, b3, (short)0, c, false, false);
  return c;
}

__device__ __forceinline__ float mlp_epilogue(v8f c, const float b1r[4], const float w2r[4],
                                              float b2s, float h[4]) {
  float z2 = b2s;
#pragma unroll
  for (int r = 0; r < 4; ++r) {
    float hr = tanhf(c[r] + b1r[r]);
    h[r] = hr;
    z2 += hr * w2r[r];
  }
  return tanhf(z2);
}

// Build A fragments (weights) once per kernel. Lane layout: lane = M (hidden row),
// VGPR0 = K + 2*hi, VGPR1 = K+1 + 2*hi within each 4-wide K chunk.
__device__ __forceinline__ void build_afrags(const float* p, int sl, int hi2, v2f a[4]) {
#pragma unroll
  for (int ck = 0; ck < 4; ++ck) {
    int kk = ck * 4 + hi2;
    float w0 = 0.f, w1 = 0.f;
    if (sl < 4) { w0 = p[sl * 16 + kk]; w1 = p[sl * 16 + kk + 1]; }
    v2f t = {w0, w1};
    a[ck] = t;
  }
}

// ---------------- init ----------------
__global__ void RenyiMI_init(const float* __restrict__ w1, const float* __restrict__ b1,
                             const float* __restrict__ w2, const float* __restrict__ b2,
                             float* __restrict__ ws) {
  int i = threadIdx.x;
  if (i < 80) {
    float v = 0.f;
    if (i < 64)      v = w1[i];
    else if (i < 68) v = b1[i - 64];
    else if (i < 72) v = w2[i - 68];
    else if (i == 72) v = b2[0];
    ws[WS_P + i] = v;
    ws[WS_M + i] = 0.f;
    ws[WS_V + i] = 0.f;
  }
}

// ---------------- pass 1: forward + per-block online logsumexp ----------------
__global__ void __launch_bounds__(NT)
RenyiMI_fwd_lse(const float* __restrict__ x, const float* __restrict__ y,
                float* __restrict__ ws, int N, unsigned pa, unsigned pc) {
  const float* p = ws + WS_P;
  const int tid = threadIdx.x;
  const int lane = tid & 31;
  const int wid = tid >> 5;
  const int hi = lane >> 4;
  const int sl = lane & 15;
  const int hi2 = hi << 1;

  v2f aw[4];
  build_afrags(p, sl, hi2, aw);
  float b1r[4], w2r[4];
#pragma unroll
  for (int r = 0; r < 4; ++r) { b1r[r] = p[64 + r]; w2r[r] = p[68 + r]; }
  const float b2s = p[72];

  float m1 = -INFINITY, s1 = 0.f, m2 = -INFINITY, s2 = 0.f;

  const long ntiles = ((long)N + 15) >> 4;
  const long nw = (long)gridDim.x * WPB;
  for (long t = (long)blockIdx.x * WPB + wid; t < ntiles; t += nw) {
    const int s0 = (int)(t << 4);
    int i = s0 + sl; if (i > N - 1) i = N - 1;
    const int j = (int)(((unsigned long long)pa * (unsigned)i + pc) % (unsigned)N);
    const float* xr = x + (size_t)i * 8;
    const float* yi = y + (size_t)i * 8;
    const float* yj = y + (size_t)j * 8;
    v2f bx0 = *(const v2f*)(xr + hi2);
    v2f bx1 = *(const v2f*)(xr + 4 + hi2);
    v2f by0 = *(const v2f*)(yi + hi2);
    v2f by1 = *(const v2f*)(yi + 4 + hi2);
    v2f bz0 = *(const v2f*)(yj + hi2);
    v2f bz1 = *(const v2f*)(yj + 4 + hi2);

    v8f acc1 = mlp_l1_wmma(aw, bx0, bx1, by0, by1);   // joint
    v8f acc2 = mlp_l1_wmma(aw, bx0, bx1, bz0, bz1);   // marginal (shuffled y)

    float h[4];
    float e1 = mlp_epilogue(acc1, b1r, w2r, b2s, h);
    float e2 = mlp_epilogue(acc2, b1r, w2r, b2s, h);

    if (hi == 0 && s0 + sl < N) {
      float f1 = (ALPHA_F - 1.f) * e1;
      float f2 = ALPHA_F * e2;
      float nm = fmaxf(m1, f1); s1 = s1 * expf(m1 - nm) + expf(f1 - nm); m1 = nm;
      nm = fmaxf(m2, f2);       s2 = s2 * expf(m2 - nm) + expf(f2 - nm); m2 = nm;
    }
  }

  warp_reduce_lse(m1, s1);
  warp_reduce_lse(m2, s2);
  __shared__ float sm1[WPB], ss1[WPB], sm2[WPB], ss2[WPB];
  if (lane == 0) { sm1[wid] = m1; ss1[wid] = s1; sm2[wid] = m2; ss2[wid] = s2; }
  __syncthreads();
  if (tid == 0) {
    float M1 = -INFINITY, S1 = 0.f, M2 = -INFINITY, S2 = 0.f;
    for (int w = 0; w < WPB; ++w) {
      lse_combine(M1, S1, sm1[w], ss1[w]);
      lse_combine(M2, S2, sm2[w], ss2[w]);
    }
    float* part = ws + WS_PART + (size_t)blockIdx.x * 4;
    part[0] = M1; part[1] = S1; part[2] = M2; part[3] = S2;
  }
}

// ---------------- pass 2: combine partials -> lse, loss; zero grads ----------------
__global__ void __launch_bounds__(256)
RenyiMI_reduce(float* __restrict__ ws, float* __restrict__ out, int nblocks, int N, int step) {
  float m1 = -INFINITY, s1 = 0.f, m2 = -INFINITY, s2 = 0.f;
  const float* part = ws + WS_PART;
  for (int b = threadIdx.x; b < nblocks; b += blockDim.x) {
    lse_combine(m1, s1, part[b * 4 + 0], part[b * 4 + 1]);
    lse_combine(m2, s2, part[b * 4 + 2], part[b * 4 + 3]);
  }
  warp_reduce_lse(m1, s1);
  warp_reduce_lse(m2, s2);
  __shared__ float sm1[8], ss1[8], sm2[8], ss2[8];
  const int lane = threadIdx.x & 31, wid = threadIdx.x >> 5;
  if (lane == 0) { sm1[wid] = m1; ss1[wid] = s1; sm2[wid] = m2; ss2[wid] = s2; }
  __syncthreads();
  if (threadIdx.x == 0) {
    float M1 = -INFINITY, S1 = 0.f, M2 = -INFINITY, S2 = 0.f;
    for (int w = 0; w < 8; ++w) {
      lse_combine(M1, S1, sm1[w], ss1[w]);
      lse_combine(M2, S2, sm2[w], ss2[w]);
    }
    float lse1 = M1 + logf(S1);
    float lse2 = M2 + logf(S2);
    ws[WS_SC] = lse1; ws[WS_SC + 1] = lse2;
    float logN = logf((float)N);
    float loss = (lse1 - logN) / (ALPHA_F - 1.f) - (lse2 - logN) / ALPHA_F;
    out[1 + step] = loss;
    if (step == NUM_STEPS - 1) out[0] = loss;
  }
  if (threadIdx.x < 80) ws[WS_G + threadIdx.x] = 0.f;
}

// ---------------- pass 3: recompute forward (WMMA) + backprop, reduce grads ----------------
__global__ void __launch_bounds__(NT)
RenyiMI_grad(const float* __restrict__ x, const float* __restrict__ y,
             float* __restrict__ ws, int N, unsigned pa, unsigned pc) {
  const float* p = ws + WS_P;
  const int tid = threadIdx.x;
  const int lane = tid & 31;
  const int wid = tid >> 5;
  const int hi = lane >> 4;
  const int sl = lane & 15;
  const int hi2 = hi << 1;

  v2f aw[4];
  build_afrags(p, sl, hi2, aw);
  float b1r[4], w2r[4];
#pragma unroll
  for (int r = 0; r < 4; ++r) { b1r[r] = p[64 + r]; w2r[r] = p[68 + r]; }
  const float b2s = p[72];
  const float lse1 = ws[WS_SC], lse2 = ws[WS_SC + 1];

  float gw1[64], gb1[4], gw2[4], gb2 = 0.f;
#pragma unroll
  for (int k = 0; k < 64; ++k) gw1[k] = 0.f;
#pragma unroll
  for (int r = 0; r < 4; ++r) { gb1[r] = 0.f; gw2[r] = 0.f; }

  const long ntiles = ((long)N + 15) >> 4;
  const long nw = (long)gridDim.x * WPB;
  for (long t = (long)blockIdx.x * WPB + wid; t < ntiles; t += nw) {
    const int s0 = (int)(t << 4);
    int i = s0 + sl; if (i > N - 1) i = N - 1;
    const int j = (int)(((unsigned long long)pa * (unsigned)i + pc) % (unsigned)N);
    const float* xr = x + (size_t)i * 8;
    const float* yi = y + (size_t)i * 8;
    const float* yj = y + (size_t)j * 8;
    v2f bx0 = *(const v2f*)(xr + hi2);
    v2f bx1 = *(const v2f*)(xr + 4 + hi2);
    v2f by0 = *(const v2f*)(yi + hi2);
    v2f by1 = *(const v2f*)(yi + 4 + hi2);
    v2f bz0 = *(const v2f*)(yj + hi2);
    v2f bz1 = *(const v2f*)(yj + 4 + hi2);

    v8f acc1 = mlp_l1_wmma(aw, bx0, bx1, by0, by1);
    v8f acc2 = mlp_l1_wmma(aw, bx0, bx1, bz0, bz1);

    float h1[4], h2[4];
    float e1 = mlp_epilogue(acc1, b1r, w2r, b2s, h1);
    float e2 = mlp_epilogue(acc2, b1r, w2r, b2s, h2);

    const bool act = (hi == 0) && (s0 + sl < N);
    if (act) {
      float inx[8], inyJ[8], inyM[8];
#pragma unroll
      for (int k = 0; k < 8; ++k) { inx[k] = xr[k]; inyJ[k] = yi[k]; inyM[k] = yj[k]; }

      // term 1 (joint): dL/de1 = -softmax(f1) = -exp((a-1)e1 - lse1)
      {
        float g = -expf((ALPHA_F - 1.f) * e1 - lse1);
        float dz2 = g * (1.f - e1 * e1);
        gb2 += dz2;
#pragma unroll
        for (int r = 0; r < 4; ++r) {
          gw2[r] += dz2 * h1[r];
          float d = dz2 * w2r[r] * (1.f - h1[r] * h1[r]);
          gb1[r] += d;
#pragma unroll
          for (int k = 0; k < 8; ++k) {
            gw1[r * 16 + k]     += d * inx[k];
            gw1[r * 16 + 8 + k] += d * inyJ[k];
          }
        }
      }
      // term 2 (marginal): dL/de2 = +exp(a*e2 - lse2)
      {
        float g = expf(ALPHA_F * e2 - lse2);
        float dz2 = g * (1.f - e2 * e2);
        gb2 += dz2;
#pragma unroll
        for (int r = 0; r < 4; ++r) {
          gw2[r] += dz2 * h2[r];
          float d = dz2 * w2r[r] * (1.f - h2[r] * h2[r]);
          gb1[r] += d;
#pragma unroll
          for (int k = 0; k < 8; ++k) {
            gw1[r * 16 + k]     += d * inx[k];
            gw1[r * 16 + 8 + k] += d * inyM[k];
          }
        }
      }
    }
  }

  // block-level gradient reduction via LDS float atomics (ds_add_f32)
  __shared__ float gacc[80];
  for (int k = tid; k < 80; k += NT) gacc[k] = 0.f;
  __syncthreads();
  if (hi == 0) {  // only lanes that ever accumulated
#pragma unroll
    for (int k = 0; k < 64; ++k) atomicAdd(&gacc[k], gw1[k]);
#pragma unroll
    for (int r = 0; r < 4; ++r) {
      atomicAdd(&gacc[64 + r], gb1[r]);
      atomicAdd(&gacc[68 + r], gw2[r]);
    }
    atomicAdd(&gacc[72], gb2);
  }
  __syncthreads();
  if (tid < 73) atomicAdd(&ws[WS_G + tid], gacc[tid]);
}

// ---------------- pass 4: Adam on 73 params ----------------
__global__ void RenyiMI_adam(float* __restrict__ ws, int t) {
  int i = threadIdx.x;
  if (i >= 73) return;
  float g = ws[WS_G + i];
  float m = ADAM_B1 * ws[WS_M + i] + (1.f - ADAM_B1) * g;
  float v = ADAM_B2 * ws[WS_V + i] + (1.f - ADAM_B2) * g * g;
  ws[WS_M + i] = m;
  ws[WS_V + i] = v;
  float bc1 = 1.f - powf(ADAM_B1, (float)t);
  float bc2 = 1.f - powf(ADAM_B2, (float)t);
  ws[WS_P + i] -= ADAM_LR * (m / bc1) / (sqrtf(v / bc2) + ADAM_EPS);
}

// ---------------- host ----------------
extern "C" void kernel_launch(void* const* d_in, const int* in_sizes, int n_in,
                              void* d_out, int out_size, void* d_ws, size_t ws_size,
                              hipStream_t stream) {
  const float* x  = (const float*)d_in[0];
  const float* y  = (const float*)d_in[1];
  const float* w1 = (const float*)d_in[2];
  const float* b1 = (const float*)d_in[3];
  const float* w2 = (const float*)d_in[4];
  const float* b2 = (const float*)d_in[5];
  float* out = (float*)d_out;
  float* ws  = (float*)d_ws;
  const int N = in_sizes[0] / 8;

  // per-step affine permutation constants: odd, not divisible by 5 -> coprime with 10^6
  static const unsigned PA[NUM_STEPS] = {123457u, 234571u, 345679u, 456791u,
                                         567899u, 678901u, 789013u, 890123u};
  static const unsigned PC[NUM_STEPS] = {97531u, 86421u, 75319u, 64207u,
                                         53197u, 42083u, 31079u, 20063u};

  RenyiMI_init<<<1, 128, 0, stream>>>(w1, b1, w2, b2, ws);
  for (int step = 0; step < NUM_STEPS; ++step) {
    RenyiMI_fwd_lse<<<NB, NT, 0, stream>>>(x, y, ws, N, PA[step], PC[step]);
    RenyiMI_reduce<<<1, 256, 0, stream>>>(ws, out, NB, N, step);
    RenyiMI_grad<<<NB, NT, 0, stream>>>(x, y, ws, N, PA[step], PC[step]);
    RenyiMI_adam<<<1, 128, 0, stream>>>(ws, step + 1);
  }
}